// MaximumMeanDiscrepancyLoss_78348793414053
// MI455X (gfx1250) — compile-verified
//
#include <hip/hip_runtime.h>

typedef __attribute__((ext_vector_type(2))) float v2f;
typedef __attribute__((ext_vector_type(8))) float v8f;

#define BATCH 4096
#define NROWS 8192
#define DIMS  256
#define NUM_K 5
#define NTILE 64                    // 8192/128 tile grid per dim
#define NTRI  (NTILE * (NTILE + 1) / 2)   // 2080 upper-triangular tiles

// ---------------- Kernel 1: row squared norms + per-block column sums -------
__global__ __launch_bounds__(256)
void mmd_norms_kernel(const float* __restrict__ s, const float* __restrict__ t,
                      float* __restrict__ sq,       // [8192]
                      float* __restrict__ colpart)  // [32*256]
{
    const int tid = threadIdx.x;
    const int row = blockIdx.x * 256 + tid;
    const float* rp = (row < BATCH) ? (s + (size_t)row * DIMS)
                                    : (t + (size_t)(row - BATCH) * DIMS);
    float acc = 0.f;
    #pragma unroll 8
    for (int k = 0; k < DIMS; k += 4) {
        float4 v = *(const float4*)(rp + k);
        acc += v.x * v.x + v.y * v.y + v.z * v.z + v.w * v.w;
    }
    sq[row] = acc;

    const int rowBase = blockIdx.x * 256;
    const float* fb = (rowBase < BATCH) ? (s + (size_t)rowBase * DIMS)
                                        : (t + (size_t)(rowBase - BATCH) * DIMS);
    float csum = 0.f;
    for (int r = 0; r < 256; ++r)
        csum += fb[(size_t)r * DIMS + tid];
    colpart[blockIdx.x * 256 + tid] = csum;
}

// ---------------- Kernel 2: bandwidth -> gamma0 -----------------------------
__global__ __launch_bounds__(256)
void mmd_gamma_kernel(const float* __restrict__ sq, const float* __restrict__ colpart,
                      float* __restrict__ g0out)
{
    __shared__ float red[256];
    const int tid = threadIdx.x;

    float a = 0.f;
    for (int k = 0; k < NROWS / 256; ++k) a += sq[tid + k * 256];
    red[tid] = a;
    __syncthreads();
    for (int st = 128; st > 0; st >>= 1) { if (tid < st) red[tid] += red[tid + st]; __syncthreads(); }
    const float sumsq = red[0];
    __syncthreads();

    float c = 0.f;
    for (int b = 0; b < 32; ++b) c += colpart[b * 256 + tid];
    red[tid] = c * c;
    __syncthreads();
    for (int st = 128; st > 0; st >>= 1) { if (tid < st) red[tid] += red[tid + st]; __syncthreads(); }

    if (tid == 0) {
        // sum(dist) = 2n*sum||x||^2 - 2*||sum x||^2   (clamp only hits ~0 diagonal)
        double sumD = 2.0 * (double)NROWS * (double)sumsq - 2.0 * (double)red[0];
        double bw   = ((double)NROWS * (double)NROWS - (double)NROWS) / sumD;
        bw *= 0.25;                      // / 2^(NUM_K//2)
        g0out[0] = (float)bw;            // gamma_i = g0 * 2^i
    }
}

// ---------------- Kernel 3: fused Gram + kernel + signed reduction ----------
#define KC 64
#define LDS_STRIDE (KC + 4)   // pad: row-to-row bank advance 4 -> conflict-free b64 frags
#define NCHUNK (DIMS / KC)

// Async L2 -> LDS copy (ASYNCcnt path); lds_off = byte offset within LDS alloc
__device__ __forceinline__ void async_b128(unsigned lds_off, const float* gptr) {
    asm volatile("global_load_async_to_lds_b128 %0, %1, off"
                 :: "v"(lds_off), "v"(gptr) : "memory");
}
__device__ __forceinline__ void wait_async0() {
    asm volatile("s_wait_asynccnt 0" ::: "memory");
}

__global__ __launch_bounds__(256)
void mmd_main_kernel(const float* __restrict__ s, const float* __restrict__ t,
                     const float* __restrict__ sq, const float* __restrict__ g0p,
                     float* __restrict__ partials)
{
    __shared__ float As[2][128 * LDS_STRIDE];
    __shared__ float Bs[2][128 * LDS_STRIDE];
    __shared__ float red[256];

    const int tid   = threadIdx.x;
    const int lane  = tid & 31;
    const int wave  = tid >> 5;          // 8 waves (wave32)
    const int waveM = wave & 1;
    const int waveN = wave >> 1;

    // decode linear block id -> upper-triangular (by, bx), bx >= by
    int rem = blockIdx.x, by = 0;
    while (rem >= NTILE - by) { rem -= NTILE - by; ++by; }
    const int bx = by + rem;

    const int rowBase = by * 128;
    const int colBase = bx * 128;
    const float* Fa = (rowBase < BATCH) ? (s + (size_t)rowBase * DIMS)
                                        : (t + (size_t)(rowBase - BATCH) * DIMS);
    const float* Fb = (colBase < BATCH) ? (s + (size_t)colBase * DIMS)
                                        : (t + (size_t)(colBase - BATCH) * DIMS);
    const float g0 = g0p[0];

    v8f zero = {};
    v8f acc[4][2];
    #pragma unroll
    for (int i = 0; i < 4; ++i)
        #pragma unroll
        for (int j = 0; j < 2; ++j) acc[i][j] = zero;

    const int lm = lane & 15;
    const int kh = (lane >> 4) * 2;

    // per-thread staging coordinates (8 x b128 per matrix per chunk)
    // slot = tid + it*256 ; r = slot/16 ; c4 = (slot%16)*4
    // prologue: stage chunk 0 into buffer 0
    #pragma unroll
    for (int it = 0; it < 8; ++it) {
        int slot = tid + it * 256;
        int r = slot >> 4, c4 = (slot & 15) << 2;
        async_b128((unsigned)(size_t)&As[0][r * LDS_STRIDE + c4], Fa + (size_t)r * DIMS + c4);
        async_b128((unsigned)(size_t)&Bs[0][r * LDS_STRIDE + c4], Fb + (size_t)r * DIMS + c4);
    }
    wait_async0();
    __syncthreads();

    for (int c = 0; c < NCHUNK; ++c) {
        const int cur = c & 1;
        // prefetch next chunk into the other buffer (overlapped with WMMA below)
        if (c + 1 < NCHUNK) {
            const int nxt = cur ^ 1;
            const int kc = (c + 1) * KC;
            #pragma unroll
            for (int it = 0; it < 8; ++it) {
                int slot = tid + it * 256;
                int r = slot >> 4, c4 = (slot & 15) << 2;
                async_b128((unsigned)(size_t)&As[nxt][r * LDS_STRIDE + c4],
                           Fa + (size_t)r * DIMS + kc + c4);
                async_b128((unsigned)(size_t)&Bs[nxt][r * LDS_STRIDE + c4],
                           Fb + (size_t)r * DIMS + kc + c4);
            }
        }

        const float* Asb = &As[cur][0];
        const float* Bsb = &Bs[cur][0];
        #pragma unroll
        for (int k0 = 0; k0 < KC; k0 += 4) {
            v2f af[4], bf[2];
            #pragma unroll
            for (int tm = 0; tm < 4; ++tm)
                af[tm] = *(const v2f*)(&Asb[(waveM * 64 + tm * 16 + lm) * LDS_STRIDE + k0 + kh]);
            #pragma unroll
            for (int tn = 0; tn < 2; ++tn)
                bf[tn] = *(const v2f*)(&Bsb[(waveN * 32 + tn * 16 + lm) * LDS_STRIDE + k0 + kh]);
            #pragma unroll
            for (int tm = 0; tm < 4; ++tm)
                #pragma unroll
                for (int tn = 0; tn < 2; ++tn)
                    acc[tm][tn] = __builtin_amdgcn_wmma_f32_16x16x4_f32(
                        false, af[tm], false, bf[tn], (short)0, acc[tm][tn], false, false);
        }

        // drain this wave's prefetch, then WG-wide handoff of buffers
        wait_async0();
        __syncthreads();
    }

    // Fused epilogue: distance -> exp chain -> signed accumulate (in regs)
    float local = 0.f;
    const int rOff = rowBase + waveM * 64;
    const int cOff = colBase + waveN * 32;
    #pragma unroll
    for (int tm = 0; tm < 4; ++tm) {
        #pragma unroll
        for (int tn = 0; tn < 2; ++tn) {
            #pragma unroll
            for (int v = 0; v < 8; ++v) {
                int row = rOff + tm * 16 + v + 8 * (lane >> 4);
                int col = cOff + tn * 16 + (lane & 15);
                float d = sq[row] + sq[col] - 2.f * acc[tm][tn][v];
                d = fmaxf(d, 0.f);
                float e1 = __expf(-d * g0);            // exp(-d*g0*2^i) = e1^(2^i)
                float e2 = e1 * e1, e4 = e2 * e2, e8 = e4 * e4, e16 = e8 * e8;
                float ks = e1 + e2 + e4 + e8 + e16;
                bool neg = (row < BATCH) != (col < BATCH);
                local += neg ? -ks : ks;
            }
        }
    }

    red[tid] = local;
    __syncthreads();
    for (int st = 128; st > 0; st >>= 1) { if (tid < st) red[tid] += red[tid + st]; __syncthreads(); }
    if (tid == 0) {
        const float w = (bx == by) ? 1.f : 2.f;   // symmetry: off-diag tiles count twice
        partials[blockIdx.x] = w * red[0];
    }
}

// ---------------- Kernel 4: deterministic final reduce ----------------------
__global__ __launch_bounds__(256)
void mmd_final_kernel(const float* __restrict__ partials, float* __restrict__ out)
{
    __shared__ float red[256];
    const int tid = threadIdx.x;
    float a = 0.f;
    for (int i = tid; i < NTRI; i += 256) a += partials[i];   // fixed order per thread
    red[tid] = a;
    __syncthreads();
    for (int st = 128; st > 0; st >>= 1) { if (tid < st) red[tid] += red[tid + st]; __syncthreads(); }
    if (tid == 0)
        out[0] = red[0] / ((float)NUM_K * (float)BATCH * (float)BATCH);
}

extern "C" void kernel_launch(void* const* d_in, const int* in_sizes, int n_in,
                              void* d_out, int out_size, void* d_ws, size_t ws_size,
                              hipStream_t stream)
{
    const float* s = (const float*)d_in[0];
    const float* t = (const float*)d_in[1];
    float* ws       = (float*)d_ws;
    float* sq       = ws;               // 8192 floats
    float* colpart  = ws + 8192;        // 8192 floats
    float* g0       = ws + 16384;       // 1 float (padded to 16)
    float* partials = ws + 16400;       // 2080 floats

    mmd_norms_kernel<<<32, 256, 0, stream>>>(s, t, sq, colpart);
    mmd_gamma_kernel<<<1, 256, 0, stream>>>(sq, colpart, g0);
    mmd_main_kernel<<<NTRI, 256, 0, stream>>>(s, t, sq, g0, partials);
    mmd_final_kernel<<<1, 256, 0, stream>>>(partials, (float*)d_out);
}